// PromptPool_32487132627376
// MI455X (gfx1250) — compile-verified
//
#include <hip/hip_runtime.h>

typedef __attribute__((ext_vector_type(2))) float v2f;
typedef __attribute__((ext_vector_type(4))) float v4f;
typedef __attribute__((ext_vector_type(8))) float v8f;

#define NUM_P   20
#define DIM     1024
#define BATCH   32768
#define TOPK    5
#define WROWS   48              // 20 norm_keys + 20 prompt_values + 8 zero pad
#define EPSN    1e-12f

// workspace layout (floats)
#define WS_W    0
#define WS_G    (WROWS * DIM)          // 49152
#define WS_H    (WS_G + 400)
#define WS_PART (WS_H + 400)
#define NTILES  (BATCH / 16)           // 2048

// output layout (floats): selected_prompts | diversity_loss | topk_indices
#define OUT_LOSS (BATCH * TOPK * DIM)
#define OUT_IDX  (OUT_LOSS + 1)

// ---------------- Phase A1: pack W = [normalize(keys); pv; zeros] ----------
__global__ void pp_prep_w(const float* __restrict__ keys,
                          const float* __restrict__ pv,
                          float* __restrict__ W) {
    int p = blockIdx.x;            // 0..47
    int t = threadIdx.x;           // 256
    __shared__ float red[256];
    float* wrow = W + (size_t)p * DIM;
    if (p < NUM_P) {
        const float* krow = keys + (size_t)p * DIM;
        float s = 0.f;
        for (int i = t; i < DIM; i += 256) { float v = krow[i]; s += v * v; }
        red[t] = s;
        __syncthreads();
        for (int off = 128; off > 0; off >>= 1) {
            if (t < off) red[t] += red[t + off];
            __syncthreads();
        }
        float inv = 1.f / fmaxf(sqrtf(red[0]), EPSN);
        for (int i = t; i < DIM; i += 256) wrow[i] = krow[i] * inv;
    } else if (p < 2 * NUM_P) {
        const float* vrow = pv + (size_t)(p - NUM_P) * DIM;
        for (int i = t; i < DIM; i += 256) wrow[i] = vrow[i];
    } else {
        for (int i = t; i < DIM; i += 256) wrow[i] = 0.f;
    }
}

// ---------------- Phase A2: Gram matrices G = pv@nk^T, H = pv@pv^T ---------
__global__ void pp_gram(const float* __restrict__ pv,
                        const float* __restrict__ W,
                        float* __restrict__ G, float* __restrict__ H) {
    int id = blockIdx.x;                 // 0..799
    int m = id / 400;                    // 0 -> G, 1 -> H
    int rem = id % 400;
    int q = rem / NUM_P, p = rem % NUM_P;
    const float* a = pv + (size_t)q * DIM;
    const float* b = (m == 0) ? (W + (size_t)p * DIM) : (pv + (size_t)p * DIM);
    __shared__ float red[256];
    int t = threadIdx.x;
    float s = 0.f;
    for (int i = t; i < DIM; i += 256) s += a[i] * b[i];
    red[t] = s;
    __syncthreads();
    for (int off = 128; off > 0; off >>= 1) {
        if (t < off) red[t] += red[t + off];
        __syncthreads();
    }
    if (t == 0) (m == 0 ? G : H)[q * NUM_P + p] = red[0];
}

// ---------------- Main: WMMA sims + selection + gather-write ---------------
__launch_bounds__(256)
__global__ void pp_main(const float* __restrict__ X,
                        const float* __restrict__ pv,
                        const float* __restrict__ Wg,
                        const float* __restrict__ Gg,
                        const float* __restrict__ Hg,
                        float* __restrict__ out_sel,
                        float* __restrict__ out_idx,
                        float* __restrict__ part) {
    __shared__ float S[8][16][48];     // per-wave sims tile [row][wcol]
    __shared__ float NSQ[8][32];       // per-lane |x|^2 partials
    __shared__ float COMM[8][16];
    __shared__ int   IDXS[8][16][TOPK];
    __shared__ float Gl[400], Hl[400];

    int tid = threadIdx.x;
    int w = tid >> 5, lane = tid & 31;
    for (int i = tid; i < 400; i += 256) { Gl[i] = Gg[i]; Hl[i] = Hg[i]; }
    __syncthreads();

    int tile  = blockIdx.x * 8 + w;
    int rbase = tile * 16;
    int mrow  = lane & 15;
    int khalf = (lane >> 4) * 2;       // K offset of this lane-half

    const float* arow = X  + (size_t)(rbase + mrow) * DIM + khalf;
    const float* b0r  = Wg + (size_t)(mrow)       * DIM + khalf;
    const float* b1r  = Wg + (size_t)(mrow + 16)  * DIM + khalf;
    const float* b2r  = Wg + (size_t)(mrow + 32)  * DIM + khalf;

    v8f c0 = {}; v8f c1 = {}; v8f c2 = {};
    float nsq = 0.f;
    #pragma unroll 4
    for (int k = 0; k < DIM; k += 4) {
        v2f a  = *(const v2f*)(arow + k);
        v2f b0 = *(const v2f*)(b0r + k);
        v2f b1 = *(const v2f*)(b1r + k);
        v2f b2 = *(const v2f*)(b2r + k);
        c0 = __builtin_amdgcn_wmma_f32_16x16x4_f32(false, a, false, b0, (short)0, c0, false, false);
        c1 = __builtin_amdgcn_wmma_f32_16x16x4_f32(false, a, false, b1, (short)0, c1, false, false);
        c2 = __builtin_amdgcn_wmma_f32_16x16x4_f32(false, a, false, b2, (short)0, c2, false, false);
        nsq = fmaf(a.x, a.x, nsq);
        nsq = fmaf(a.y, a.y, nsq);
    }

    // C layout: VGPR v -> (M = v + 8*(lane>=16), N = lane&15)
    int crow = (lane >> 4) * 8;
    int ccol = lane & 15;
    #pragma unroll
    for (int v = 0; v < 8; ++v) {
        S[w][crow + v][ccol]      = c0[v];
        S[w][crow + v][16 + ccol] = c1[v];
        S[w][crow + v][32 + ccol] = c2[v];
    }
    NSQ[w][lane] = nsq;
    __syncthreads();

    if (lane < 16) {
        int r = lane;
        int grow = rbase + r;
        float d[NUM_P], rv[NUM_P];
        #pragma unroll
        for (int p = 0; p < NUM_P; ++p) {
            d[p]  = S[w][r][p];            // dot(r, nk_p)
            rv[p] = S[w][r][NUM_P + p];    // dot(r, pv_p)
        }
        float rn2 = NSQ[w][r] + NSQ[w][r + 16];
        float comm = 0.f;
        #pragma unroll
        for (int t = 0; t < TOPK; ++t) {
            float inv = 1.f / fmaxf(sqrtf(fmaxf(rn2, 0.f)), EPSN);
            float best = d[0]; int bi = 0;
            #pragma unroll
            for (int p = 1; p < NUM_P; ++p) {
                if (d[p] > best) { best = d[p]; bi = p; }  // first-occurrence argmax
            }
            float mx = best * inv;
            float se = 0.f;
            #pragma unroll
            for (int p = 0; p < NUM_P; ++p) se += expf(d[p] * inv - mx);
            comm += 1.f / se;              // softmax prob of the argmax
            IDXS[w][r][t] = bi;
            out_idx[(size_t)grow * TOPK + t] = (float)bi;
            // residual update in dot-space
            rn2 = rn2 - 2.f * rv[bi] + Hl[bi * NUM_P + bi];
            #pragma unroll
            for (int p = 0; p < NUM_P; ++p) {
                d[p]  -= Gl[bi * NUM_P + p];
                rv[p] -= Hl[bi * NUM_P + p];
            }
        }
        COMM[w][r] = comm;
    }
    __syncthreads();

    if (lane == 0) {                      // deterministic per-tile reduction
        float s = 0.f;
        for (int r = 0; r < 16; ++r) s += COMM[w][r];
        part[tile] = s;
    }

    // gather-write selected prompts: pv rows are L2/WGP$-hot, stores NT
    for (int rt = 0; rt < 16 * TOPK; ++rt) {
        int r = rt / TOPK, t = rt % TOPK;
        int pi = IDXS[w][r][t];
        const float* src = pv + (size_t)pi * DIM;
        float* dst = out_sel + ((size_t)(rbase + r) * TOPK + t) * DIM;
        for (int j = lane; j < DIM / 4; j += 32) {
            v4f val = *(const v4f*)(src + j * 4);
            __builtin_nontemporal_store(val, (v4f*)(dst + j * 4));
        }
    }
}

// ---------------- Final: deterministic loss reduction ----------------------
__global__ void pp_loss(const float* __restrict__ part, float* __restrict__ out_loss) {
    if (threadIdx.x == 0 && blockIdx.x == 0) {
        float s = 0.f;
        for (int i = 0; i < NTILES; ++i) s += part[i];
        out_loss[0] = -s / (float)BATCH;
    }
}

extern "C" void kernel_launch(void* const* d_in, const int* in_sizes, int n_in,
                              void* d_out, int out_size, void* d_ws, size_t ws_size,
                              hipStream_t stream) {
    const float* X    = (const float*)d_in[0];   // [B, D]
    const float* keys = (const float*)d_in[1];   // [P, D]
    const float* pv   = (const float*)d_in[2];   // [P, D]
    // d_in[3] = top_k (==5), compiled in.

    float* ws   = (float*)d_ws;
    float* W    = ws + WS_W;
    float* G    = ws + WS_G;
    float* H    = ws + WS_H;
    float* part = ws + WS_PART;
    float* out  = (float*)d_out;

    pp_prep_w<<<WROWS, 256, 0, stream>>>(keys, pv, W);
    pp_gram<<<800, 256, 0, stream>>>(pv, W, G, H);
    pp_main<<<BATCH / 128, 256, 0, stream>>>(X, pv, W, G, H,
                                             out, out + OUT_IDX, part);
    pp_loss<<<1, 32, 0, stream>>>(part, out + OUT_LOSS);
}